// TiedCoxLoss_39204461478243
// MI455X (gfx1250) — compile-verified
//
#include <hip/hip_runtime.h>
#include <hip/hip_bf16.h>

#define T_BUCKETS 8192
#define HIST_BLOCKS 256
#define HIST_THREADS 256

// ws layout (all T_BUCKETS entries, 4B each):
//   [0]      S_g   float : sum exp(pred) per bucket
//   [8192]   P_g   float : sum pred per bucket
//   [16384]  C_g   uint  : (count << 1) | any_observed_bit
//   [24576]  R_g   float : suffix sum of S (risk set)

// ---------------- Pass 1: LDS-privatized bucket histogram ----------------
// Per-block LDS packing of the count word:
//   bits [16:1]  = tie count (block-local, max ~31.3K elements/block)
//   bits [31:17] = observed count (block-local)
// -> one unconditional 32-bit LDS atomicAdd covers both, no divergent atomicOr.
__global__ __launch_bounds__(HIST_THREADS)
void cox_hist_kernel(const float* __restrict__ preds,
                     const int* __restrict__ times,
                     const int* __restrict__ observed,
                     float* __restrict__ S_g,
                     float* __restrict__ P_g,
                     unsigned* __restrict__ C_g,
                     int n)
{
    // 96 KB of LDS per workgroup — feasible on CDNA5's 320 KB WGP LDS,
    // still allows ~3 concurrent workgroups per WGP.
    __shared__ float    sS[T_BUCKETS];
    __shared__ float    sP[T_BUCKETS];
    __shared__ unsigned sC[T_BUCKETS];

    for (int b = threadIdx.x; b < T_BUCKETS; b += HIST_THREADS) {
        sS[b] = 0.0f; sP[b] = 0.0f; sC[b] = 0u;
    }
    __syncthreads();

    const int stride = gridDim.x * HIST_THREADS;
    const int gid    = blockIdx.x * HIST_THREADS + threadIdx.x;
    const int nv     = n >> 2;   // vectorized element count

    const float4* __restrict__ p4 = (const float4*)preds;
    const int4*   __restrict__ t4 = (const int4*)times;
    const int4*   __restrict__ o4 = (const int4*)observed;

    for (int i = gid; i < nv; i += stride) {
        if (i + stride < nv) {
            __builtin_prefetch(&p4[i + stride], 0, 1);  // global_prefetch_b8
        }
        float4 p = p4[i];
        int4   t = t4[i];
        int4   o = o4[i];

        {   float e = __expf(p.x); atomicAdd(&sS[t.x], e); atomicAdd(&sP[t.x], p.x);
            atomicAdd(&sC[t.x], 2u + ((unsigned)o.x << 17)); }
        {   float e = __expf(p.y); atomicAdd(&sS[t.y], e); atomicAdd(&sP[t.y], p.y);
            atomicAdd(&sC[t.y], 2u + ((unsigned)o.y << 17)); }
        {   float e = __expf(p.z); atomicAdd(&sS[t.z], e); atomicAdd(&sP[t.z], p.z);
            atomicAdd(&sC[t.z], 2u + ((unsigned)o.z << 17)); }
        {   float e = __expf(p.w); atomicAdd(&sS[t.w], e); atomicAdd(&sP[t.w], p.w);
            atomicAdd(&sC[t.w], 2u + ((unsigned)o.w << 17)); }
    }
    // scalar tail (N % 4)
    for (int i = (nv << 2) + gid; i < n; i += stride) {
        float p = preds[i];
        int   t = times[i];
        float e = __expf(p);
        atomicAdd(&sS[t], e); atomicAdd(&sP[t], p);
        atomicAdd(&sC[t], 2u + ((unsigned)observed[i] << 17));
    }
    __syncthreads();

    // flush privatized bins to global accumulators
    for (int b = threadIdx.x; b < T_BUCKETS; b += HIST_THREADS) {
        unsigned c = sC[b];
        if (c) {
            unsafeAtomicAdd(&S_g[b], sS[b]);            // global_atomic_add_f32
            unsafeAtomicAdd(&P_g[b], sP[b]);
            unsigned cnt_local = (c >> 1) & 0xFFFFu;    // tie count
            unsigned obs_local = c >> 17;               // observed count
            atomicAdd(&C_g[b], cnt_local << 1);
            if (obs_local) atomicOr(&C_g[b], 1u);
        }
    }
}

// ---------------- Pass 2: suffix sum R[t] = sum_{t' >= t} S[t'] ----------------
__global__ __launch_bounds__(1024)
void cox_scan_kernel(const float* __restrict__ S_g, float* __restrict__ R_g)
{
    __shared__ float sS[T_BUCKETS];
    __shared__ float ts[1024];

    const int i = threadIdx.x;

    // Bulk-copy S_g (32 KB) global -> LDS with CDNA5 async-to-LDS loads:
    // each lane moves two 16B chunks, tracked by ASYNCcnt (no VGPR staging).
    {
        unsigned lds_base =
            (unsigned)(uintptr_t)((__attribute__((address_space(3))) float*)sS);
        unsigned l0 = lds_base + (unsigned)i * 16u;          // bytes [i*16 .. )
        unsigned l1 = l0 + 16384u;                           // second half
        const float* g0 = S_g + i * 4;                       // 4 floats = 16B
        const float* g1 = g0 + 4096;
        asm volatile("global_load_async_to_lds_b128 %0, %1, off"
                     :: "v"(l0), "v"(g0) : "memory");
        asm volatile("global_load_async_to_lds_b128 %0, %1, off"
                     :: "v"(l1), "v"(g1) : "memory");
        asm volatile("s_wait_asynccnt 0" ::: "memory");
    }
    __syncthreads();

    // each thread owns 8 contiguous buckets
    float tsum = 0.0f;
    #pragma unroll
    for (int k = 0; k < 8; ++k) tsum += sS[i * 8 + k];
    ts[i] = tsum;
    __syncthreads();

    // Hillis-Steele inclusive scan toward lower indices:
    // after loop, ts[i] = sum_{j >= i} tsum_j
    for (int off = 1; off < 1024; off <<= 1) {
        float v = (i + off < 1024) ? ts[i + off] : 0.0f;
        __syncthreads();
        ts[i] += v;
        __syncthreads();
    }

    float run = (i < 1023) ? ts[i + 1] : 0.0f;   // sum over all later threads
    #pragma unroll
    for (int k = 7; k >= 0; --k) {
        run += sS[i * 8 + k];
        R_g[i * 8 + k] = run;
    }
}

// ---------------- Pass 3: Efron log terms, one wave32 per bucket ----------------
__global__ __launch_bounds__(256)
void cox_efron_kernel(const float* __restrict__ S_g,
                      const float* __restrict__ P_g,
                      const unsigned* __restrict__ C_g,
                      const float* __restrict__ R_g,
                      float* __restrict__ out)
{
    const int wave = threadIdx.x >> 5;
    const int lane = threadIdx.x & 31;
    const int t    = blockIdx.x * 8 + wave;   // grid = T_BUCKETS/8 blocks

    unsigned c = C_g[t];
    unsigned m = c >> 1;
    if ((c & 1u) && m > 0u) {
        float R   = R_g[t];
        float S   = S_g[t];
        float inv = 1.0f / (float)m;
        float part = 0.0f;
        for (unsigned l = lane; l < m; l += 32u) {
            part += __logf(R - (float)l * inv * S);
        }
        // wave32 butterfly reduction
        #pragma unroll
        for (int off = 16; off > 0; off >>= 1)
            part += __shfl_xor(part, off, 32);

        if (lane == 0) {
            // loss = sum(log terms) - sum(obs * P)
            unsafeAtomicAdd(out, part - P_g[t]);
        }
    }
}

extern "C" void kernel_launch(void* const* d_in, const int* in_sizes, int n_in,
                              void* d_out, int out_size, void* d_ws, size_t ws_size,
                              hipStream_t stream) {
    const float* preds    = (const float*)d_in[0];
    const int*   times    = (const int*)d_in[1];
    const int*   observed = (const int*)d_in[2];
    const int    n        = in_sizes[0];

    float*    S_g = (float*)d_ws;
    float*    P_g = S_g + T_BUCKETS;
    unsigned* C_g = (unsigned*)(P_g + T_BUCKETS);
    float*    R_g = (float*)(C_g + T_BUCKETS);
    float*    out = (float*)d_out;

    // zero accumulators every call (graph-capturable)
    hipMemsetAsync(d_ws, 0, 4u * T_BUCKETS * sizeof(float), stream);
    hipMemsetAsync(d_out, 0, sizeof(float), stream);

    cox_hist_kernel<<<HIST_BLOCKS, HIST_THREADS, 0, stream>>>(
        preds, times, observed, S_g, P_g, C_g, n);
    cox_scan_kernel<<<1, 1024, 0, stream>>>(S_g, R_g);
    cox_efron_kernel<<<T_BUCKETS / 8, 256, 0, stream>>>(S_g, P_g, C_g, R_g, out);
}